// SimpleGCN_82583631168190
// MI455X (gfx1250) — compile-verified
//
#include <hip/hip_runtime.h>

typedef __attribute__((ext_vector_type(2))) float v2f;
typedef __attribute__((ext_vector_type(8))) float v8f;

static constexpr int NN = 100000;  // nodes
static constexpr int FI = 128;     // in features
static constexpr int FH = 32;      // hidden features

// ---------------- init: zero accumulators, deg = 1 (self loop), out = 0 -------
__global__ void k_init(float* __restrict__ deg, float* __restrict__ out1,
                       float* __restrict__ out2, float* __restrict__ out) {
  int i = blockIdx.x * blockDim.x + threadIdx.x;
  if (i < NN * FH) out1[i] = 0.0f;
  if (i < NN) { deg[i] = 1.0f; out2[i] = 0.0f; }
  if (i == 0) out[0] = 0.0f;
}

// ---------------- degree accumulation over dst ---------------------------------
__global__ void k_degree(const long long* __restrict__ dst, int E,
                         float* __restrict__ deg) {
  int e = blockIdx.x * blockDim.x + threadIdx.x;
  if (e < E) atomicAdd(&deg[(int)dst[e]], 1.0f);
}

// ---------------- deg -> rsqrt(deg) in place -----------------------------------
__global__ void k_dinv(float* __restrict__ deg) {
  int i = blockIdx.x * blockDim.x + threadIdx.x;
  if (i < NN) {
    float d = deg[i];
    deg[i] = (d > 0.0f) ? rsqrtf(d) : 0.0f;
  }
}

// ---------------- h1 = x @ W1 via V_WMMA_F32_16X16X4_F32 -----------------------
// 256 threads = 8 waves; each wave computes a 16x32 tile of h1.
// W1 staged in LDS pre-packed as K-pair float2 so every B fragment is one
// 8-byte ds_load_b64 into an even-aligned VGPR pair (no repack moves).
__global__ void __launch_bounds__(256) k_gemm1(const float* __restrict__ x,
                                               const float* __restrict__ W1,
                                               float* __restrict__ h1) {
  __shared__ v2f pw[(FI / 2) * FH];  // 64 K-pairs x 32 cols = 16 KB
  for (int i = threadIdx.x; i < (FI / 2) * FH; i += 256) {
    int k2 = i >> 5;   // K-pair index (K = 2*k2, 2*k2+1)
    int n  = i & 31;   // output column
    v2f t;
    t.x = W1[(2 * k2) * FH + n];
    t.y = W1[(2 * k2 + 1) * FH + n];
    pw[i] = t;
  }
  __syncthreads();

  const int wave = threadIdx.x >> 5;
  const int lane = threadIdx.x & 31;
  const int half = lane >> 4;   // 0: lanes 0-15, 1: lanes 16-31
  const int l16  = lane & 15;
  const int m0   = (blockIdx.x * 8 + wave) * 16;
  if (m0 >= NN) return;         // whole-wave uniform branch, EXEC stays all-1s

  // A fragment source: row m = m0+l16, K pair starts at 2*half (ISA 7.12.2)
  const float* arow = x + (size_t)(m0 + l16) * FI + 2 * half;
  __builtin_prefetch(arow + 64, 0, 0);   // global_prefetch_b8: 2nd half of row

  v8f c0 = {};  // cols 0..15
  v8f c1 = {};  // cols 16..31
  for (int k0 = 0; k0 < FI; k0 += 4) {
    v2f a = *(const v2f*)(arow + k0);    // contiguous, 8B aligned
    const int kp = (k0 >> 1) + half;     // lane's K-pair: K = k0+2*half, +1
    v2f b0 = pw[kp * FH + l16];          // ds_load_b64
    v2f b1 = pw[kp * FH + 16 + l16];     // ds_load_b64
    c0 = __builtin_amdgcn_wmma_f32_16x16x4_f32(false, a, false, b0, (short)0, c0,
                                               false, false);
    c1 = __builtin_amdgcn_wmma_f32_16x16x4_f32(false, a, false, b1, (short)0, c1,
                                               false, false);
  }

  // C/D layout: lane l -> col l%16, rows v + 8*(l/16) for VGPR v
  float* hrow = h1 + (size_t)m0 * FH;
#pragma unroll
  for (int v = 0; v < 8; ++v) {
    int m = v + 8 * half;
    hrow[m * FH + l16]      = c0[v];
    hrow[m * FH + 16 + l16] = c1[v];
  }
}

// ---------------- edge scatter layer 1: one wave per edge, lane = feature ------
__global__ void k_scatter1(const long long* __restrict__ src,
                           const long long* __restrict__ dst, int E,
                           const float* __restrict__ dinv,
                           const float* __restrict__ h1,
                           float* __restrict__ out1) {
  int gid = blockIdx.x * blockDim.x + threadIdx.x;
  int e = gid >> 5;
  int f = gid & 31;
  if (e < E) {
    int s = (int)src[e];
    int d = (int)dst[e];
    float norm = dinv[s] * dinv[d];
    atomicAdd(&out1[(size_t)d * FH + f], h1[(size_t)s * FH + f] * norm);
  }
}

// ------- fused: self-loop + bias + ReLU + (out1 @ W2) per-node dot -------------
// one wave per node, lane = feature; avoids materializing the activated out1.
__global__ void k_fuse12(const float* __restrict__ out1,
                         const float* __restrict__ h1,
                         const float* __restrict__ dinv,
                         const float* __restrict__ b1,
                         const float* __restrict__ W2,
                         float* __restrict__ h2) {
  int gid = blockIdx.x * blockDim.x + threadIdx.x;
  int i = gid >> 5;
  int f = gid & 31;
  if (i < NN) {
    float di = dinv[i];
    size_t idx = (size_t)i * FH + f;
    float v = out1[idx] + h1[idx] * di * di + b1[f];  // segment_sum + self + bias
    v = v > 0.0f ? v : 0.0f;                          // ReLU
    v *= W2[f];                                       // layer-2 linear (32 -> 1)
#pragma unroll
    for (int o = 16; o > 0; o >>= 1) v += __shfl_xor(v, o, 32);
    if (f == 0) h2[i] = v;
  }
}

// ---------------- edge scatter layer 2 (1 float per edge) ----------------------
__global__ void k_scatter2(const long long* __restrict__ src,
                           const long long* __restrict__ dst, int E,
                           const float* __restrict__ dinv,
                           const float* __restrict__ h2,
                           float* __restrict__ out2) {
  int e = blockIdx.x * blockDim.x + threadIdx.x;
  if (e < E) {
    int s = (int)src[e];
    int d = (int)dst[e];
    atomicAdd(&out2[d], h2[s] * dinv[s] * dinv[d]);
  }
}

// ---------------- self-loop + bias + mean over nodes ---------------------------
__global__ void k_final2(const float* __restrict__ out2,
                         const float* __restrict__ h2,
                         const float* __restrict__ dinv,
                         const float* __restrict__ b2,
                         float* __restrict__ out) {
  int i = blockIdx.x * blockDim.x + threadIdx.x;
  float v = 0.0f;
  if (i < NN) {
    float di = dinv[i];
    v = (out2[i] + h2[i] * di * di + b2[0]) * (1.0f / (float)NN);
  }
#pragma unroll
  for (int o = 16; o > 0; o >>= 1) v += __shfl_xor(v, o, 32);
  if ((threadIdx.x & 31) == 0) atomicAdd(out, v);
}

extern "C" void kernel_launch(void* const* d_in, const int* in_sizes, int n_in,
                              void* d_out, int out_size, void* d_ws, size_t ws_size,
                              hipStream_t stream) {
  const float* x        = (const float*)d_in[0];
  const long long* eidx = (const long long*)d_in[1];  // (2, E) int64
  const float* W1       = (const float*)d_in[2];
  const float* b1       = (const float*)d_in[3];
  const float* W2       = (const float*)d_in[4];
  const float* b2       = (const float*)d_in[5];
  float* out            = (float*)d_out;

  const int E = in_sizes[1] / 2;
  const long long* src = eidx;
  const long long* dst = eidx + E;

  // workspace layout (floats): deg/dinv | h1 | out1 | h2 | out2  (~27 MB)
  float* ws   = (float*)d_ws;
  float* deg  = ws;
  float* h1   = deg + NN;
  float* out1 = h1 + (size_t)NN * FH;
  float* h2   = out1 + (size_t)NN * FH;
  float* out2 = h2 + NN;

  const int nelem = NN * FH;
  k_init<<<(nelem + 255) / 256, 256, 0, stream>>>(deg, out1, out2, out);
  k_degree<<<(E + 255) / 256, 256, 0, stream>>>(dst, E, deg);
  k_dinv<<<(NN + 255) / 256, 256, 0, stream>>>(deg);
  k_gemm1<<<((NN / 16) + 7) / 8, 256, 0, stream>>>(x, W1, h1);
  k_scatter1<<<(E * 32 + 255) / 256, 256, 0, stream>>>(src, dst, E, deg, h1, out1);
  k_fuse12<<<(NN * 32 + 255) / 256, 256, 0, stream>>>(out1, h1, deg, b1, W2, h2);
  k_scatter2<<<(E + 255) / 256, 256, 0, stream>>>(src, dst, E, deg, h2, out2);
  k_final2<<<(NN + 255) / 256, 256, 0, stream>>>(out2, h2, deg, b2, out);
}